// AttentionCDE_21835613733136
// MI455X (gfx1250) — compile-verified
//
#include <hip/hip_runtime.h>
#include <math.h>

typedef float v2f __attribute__((ext_vector_type(2)));
typedef float v8f __attribute__((ext_vector_type(8)));

#define FEAT      150528
#define S_STEPS   190
#define KCHUNK    2352
#define NCHUNKS   64
#define NWG_SCAN  64

// workspace layout (float offsets)
#define WS_HPART  0         // 64 * 8192 partial backbone sums
#define WS_X      524288    // 64*64
#define WS_A0     528384    // 16*16
#define WS_Z      528640    // 128*16
#define WS_EV     530688    // 190*64
#define WS_DATT   542848    // 190*16
#define WS_BAR    545888    // 2 uints

__device__ __forceinline__ v8f wmma_f32_4(v2f a, v2f b, v8f c) {
  // D = A(16x4 f32) * B(4x16 f32) + C(16x16 f32)
  return __builtin_amdgcn_wmma_f32_16x16x4_f32(false, a, false, b, (short)0, c,
                                               false, false);
}

// ---------------------------------------------------------------------------
// K1: backbone GEMM  frames[64,150528] @ bb_w1[150528,128], split-K partials.
// One wave per 16x16 output tile per K-chunk. Memory-bound: streams A/B once.
// ---------------------------------------------------------------------------
__global__ __launch_bounds__(32) void k_bb(const float* __restrict__ A,
                                           const float* __restrict__ B,
                                           float* __restrict__ hpart) {
  const int m0 = blockIdx.x * 16, n0 = blockIdx.y * 16;
  const long kbase = (long)blockIdx.z * KCHUNK;
  const int lane = threadIdx.x;
  const int hf = lane >> 4, r = lane & 15;
  const float* ap = A + (size_t)(m0 + r) * FEAT + kbase + 2 * hf;
  const float* bp = B + (size_t)(kbase + 2 * hf) * 128 + n0 + r;
  v8f c = {};
  for (int k = 0; k < KCHUNK; k += 4) {
    v2f a; a.x = ap[0]; a.y = ap[1];
    v2f b; b.x = bp[0]; b.y = bp[128];
    c = wmma_f32_4(a, b, c);
    ap += 4; bp += 4 * 128;
  }
  float* o = hpart + (size_t)blockIdx.z * 8192;
#pragma unroll
  for (int i = 0; i < 8; ++i)
    o[(m0 + i + 8 * hf) * 128 + (n0 + r)] = c[i];
}

// ---------------------------------------------------------------------------
// K2: reduce partials -> hmid (bias+relu), x = hmid@bb_w2+b2,
//     attention row 0 -> a0, z0 = einsum(g_h_encoder, x[0]).T
// ---------------------------------------------------------------------------
__global__ __launch_bounds__(256) void k_mid(
    const float* __restrict__ hpart, const float* __restrict__ bb_b1,
    const float* __restrict__ bb_w2, const float* __restrict__ bb_b2,
    const float* __restrict__ key_w, const float* __restrict__ key_b,
    const float* __restrict__ query_w, const float* __restrict__ query_b,
    const float* __restrict__ value_w, const float* __restrict__ value_b,
    const float* __restrict__ ghe,
    float* __restrict__ xg, float* __restrict__ a0g, float* __restrict__ zbuf) {
  __shared__ float sh[64][128];
  __shared__ float sx[64][64];
  __shared__ float sq0[256];
  __shared__ float sp[64];
  const int tid = threadIdx.x;
  // hmid = relu(sum_chunks + b1)
  for (int idx = tid; idx < 8192; idx += 256) {
    float acc = bb_b1[idx & 127];
    for (int c = 0; c < NCHUNKS; ++c) acc += hpart[(size_t)c * 8192 + idx];
    sh[idx >> 7][idx & 127] = acc > 0.f ? acc : 0.f;
  }
  __syncthreads();
  // x = hmid @ bb_w2 + b2
  for (int idx = tid; idx < 4096; idx += 256) {
    int t = idx >> 6, d = idx & 63;
    float acc = bb_b2[d];
    for (int j = 0; j < 128; ++j) acc += sh[t][j] * bb_w2[j * 64 + d];
    sx[t][d] = acc; xg[idx] = acc;
  }
  __syncthreads();
  // q0 = x[0] @ query_w + qb
  {
    int j = tid;
    float acc = query_b[j];
    for (int d = 0; d < 64; ++d) acc += sx[0][d] * query_w[d * 256 + j];
    sq0[j] = acc;
  }
  __syncthreads();
  // logits[t] = q0 . k[t] / 16
  if (tid < 64) {
    int t = tid; float acc = 0.f;
    for (int j = 0; j < 256; ++j) {
      float kj = key_b[j];
      for (int d = 0; d < 64; ++d) kj += sx[t][d] * key_w[d * 256 + j];
      acc += sq0[j] * kj;
    }
    sp[t] = acc * (1.0f / 16.0f);
  }
  __syncthreads();
  if (tid == 0) {  // softmax over 64 (cheap, deterministic)
    float mx = sp[0];
    for (int t = 1; t < 64; ++t) mx = fmaxf(mx, sp[t]);
    float sum = 0.f;
    for (int t = 0; t < 64; ++t) { float e = expf(sp[t] - mx); sp[t] = e; sum += e; }
    float inv = 1.f / sum;
    for (int t = 0; t < 64; ++t) sp[t] *= inv;
  }
  __syncthreads();
  // a0[j] = sum_t p[t] * v[t][j]
  {
    int j = tid; float acc = 0.f;
    for (int t = 0; t < 64; ++t) {
      float v = value_b[j];
      for (int d = 0; d < 64; ++d) v += sx[t][d] * value_w[d * 256 + j];
      acc += sp[t] * v;
    }
    a0g[j] = acc;
  }
  // z0[h][n] = sum_d ghe[n][h][d] * x[0][d]
  for (int idx = tid; idx < 2048; idx += 256) {
    int h = idx >> 4, n = idx & 15;
    float acc = 0.f;
    for (int d = 0; d < 64; ++d) acc += ghe[n * 8192 + h * 64 + d] * sx[0][d];
    zbuf[idx] = acc;
  }
}

// ---------------------------------------------------------------------------
// K3: cubic spline (Thomas solve, uniform spacing) -> evals[190,64], datt[190,16]
// ---------------------------------------------------------------------------
__global__ __launch_bounds__(256) void k_spline(
    const float* __restrict__ xg, const float* __restrict__ xd_w,
    const float* __restrict__ xd_b, float* __restrict__ evg,
    float* __restrict__ dag) {
  __shared__ float snd[64];
  __shared__ float snb[64][64];
  const int tid = threadIdx.x;
  if (tid == 0) {  // pivots shared across all D columns (uniform spacing)
    snd[0] = 2.f;
    for (int i = 1; i < 63; ++i) snd[i] = 4.f - 1.f / snd[i - 1];
    snd[63] = 2.f - 1.f / snd[62];
  }
  __syncthreads();
  if (tid < 64) {
    int d = tid;
    float prev = 3.f * (xg[64 + d] - xg[d]);
    snb[d][0] = prev;
    for (int i = 1; i < 64; ++i) {
      float rhs;
      if (i < 63) {
        float pim1 = 3.f * (xg[i * 64 + d] - xg[(i - 1) * 64 + d]);
        float pi   = 3.f * (xg[(i + 1) * 64 + d] - xg[i * 64 + d]);
        rhs = pim1 + pi;
      } else {
        rhs = 3.f * (xg[63 * 64 + d] - xg[62 * 64 + d]);
      }
      prev = rhs - prev / snd[i - 1];
      snb[d][i] = prev;
    }
    float nxt = snb[d][63] / snd[63];
    snb[d][63] = nxt;
    for (int i = 62; i >= 0; --i) {
      nxt = (snb[d][i] - nxt) / snd[i];
      snb[d][i] = nxt;  // kd
    }
  }
  __syncthreads();
  for (int idx = tid; idx < S_STEPS * 64; idx += 256) {
    int s = idx >> 6, d = idx & 63;
    int i0 = (s == 0) ? 0 : (s - 1) / 3;
    float frac = (float)s * (1.f / 3.f) - (float)i0;
    float p0 = xg[i0 * 64 + d], p1 = xg[(i0 + 1) * 64 + d];
    float pd = p1 - p0;
    float k0 = snb[d][i0], k1 = snb[d][i0 + 1];
    float twoc = 6.f * pd - 4.f * k0 - 2.f * k1;
    float threed = -6.f * pd + 3.f * (k0 + k1);
    evg[idx] = p0 + (k0 + (0.5f * twoc + threed * frac * (1.f / 3.f)) * frac) * frac;
  }
  __threadfence();
  __syncthreads();
  for (int idx = tid; idx < S_STEPS * 16; idx += 256) {
    int s = idx >> 4, j = idx & 15;
    float acc = xd_b[j];
    for (int d = 0; d < 64; ++d) acc += evg[s * 64 + d] * xd_w[d * 16 + j];
    dag[idx] = acc;
  }
}

// ---------------------------------------------------------------------------
// sense-reversing device-wide barrier (agent-scope atomics)
// ---------------------------------------------------------------------------
__device__ __forceinline__ void gbarrier(unsigned* cnt, unsigned* gen) {
  __syncthreads();
  if (threadIdx.x == 0) {
    unsigned cur = __hip_atomic_load(gen, __ATOMIC_RELAXED, __HIP_MEMORY_SCOPE_AGENT);
    unsigned got = __hip_atomic_fetch_add(cnt, 1u, __ATOMIC_ACQ_REL, __HIP_MEMORY_SCOPE_AGENT);
    if (got == NWG_SCAN - 1) {
      __hip_atomic_store(cnt, 0u, __ATOMIC_RELAXED, __HIP_MEMORY_SCOPE_AGENT);
      __hip_atomic_fetch_add(gen, 1u, __ATOMIC_ACQ_REL, __HIP_MEMORY_SCOPE_AGENT);
    } else {
      while (__hip_atomic_load(gen, __ATOMIC_ACQUIRE, __HIP_MEMORY_SCOPE_AGENT) == cur)
        __builtin_amdgcn_s_sleep(2);
    }
  }
  __syncthreads();
}

// ---------------------------------------------------------------------------
// K4: persistent scan. 64 WGs, WG g owns h rows {2g,2g+1} (=128 contiguous
// columns of cde_w2, L2-resident). Two WMMA GEMMs per step, z exchanged via
// global zbuf with two device barriers per step.
// ---------------------------------------------------------------------------
__global__ __launch_bounds__(128) void k_scan(
    const float* __restrict__ w1, const float* __restrict__ b1g,
    const float* __restrict__ w2, const float* __restrict__ b2g,
    const float* __restrict__ zta,
    const float* __restrict__ aw1, const float* __restrict__ ab1,
    const float* __restrict__ aw2, const float* __restrict__ ab2,
    const float* __restrict__ evg, const float* __restrict__ dag,
    const float* __restrict__ a0g,
    float* __restrict__ zbuf, unsigned* bar) {
  __shared__ float sz[128][16];   // full z (h,n)
  __shared__ float sZ[16][128];   // Z = (z@a)^T
  __shared__ float sH[16][128];   // h1
  __shared__ float sG[16][128];   // tanh(h1@w2slice+b2)
  __shared__ float sa[16][16];
  __shared__ float sda[16][16];
  __shared__ float shid[16][32];
  __shared__ float sb1[128], sb2[128];
  __shared__ float sw1a[512], sw2a[512], sab1[32], sab2[16];
  __shared__ float sxe[64], sde[16];
  const int tid = threadIdx.x;
  const int g = blockIdx.x;
  const int gb = g * 128;  // column base into cde_w2 / cde_b2

  for (int i = tid; i < 2048; i += 128) sz[i >> 4][i & 15] = zbuf[i];
  for (int i = tid; i < 256; i += 128) sa[i >> 4][i & 15] = a0g[i];
  sb1[tid] = b1g[tid];
  sb2[tid] = b2g[gb + tid];
  for (int i = tid; i < 512; i += 128) { sw1a[i] = aw1[i]; sw2a[i] = aw2[i]; }
  if (tid < 32) sab1[tid] = ab1[tid];
  if (tid < 16) sab2[tid] = ab2[tid];
  __syncthreads();

  const int lane = tid & 31, wv = tid >> 5;
  const int hf = lane >> 4, r = lane & 15;

  for (int s = 0; s < S_STEPS; ++s) {
    if (tid < 64) sxe[tid] = evg[s * 64 + tid];
    if (tid < 16) sde[tid] = dag[s * 16 + tid];
    __syncthreads();
    // Z[n][h] = sum_m z[h][m] * a[m][n]
    {
      int h = tid;
      for (int n = 0; n < 16; ++n) {
        float acc = 0.f;
        for (int m = 0; m < 16; ++m) acc += sz[h][m] * sa[m][n];
        sZ[n][h] = acc;
      }
    }
    __syncthreads();
    // GEMM1: h1 = relu(Z @ w1 + b1)   [16,128]x[128,128]
    for (int nt = 2 * wv; nt < 2 * wv + 2; ++nt) {
      int n0 = nt * 16;
      v8f c = {};
      for (int kk = 0; kk < 128; kk += 4) {
        v2f a; a.x = sZ[r][kk + 2 * hf]; a.y = sZ[r][kk + 2 * hf + 1];
        const float* bp = w1 + (kk + 2 * hf) * 128 + n0 + r;
        v2f b; b.x = bp[0]; b.y = bp[128];
        c = wmma_f32_4(a, b, c);
      }
#pragma unroll
      for (int i = 0; i < 8; ++i) {
        float v = c[i] + sb1[n0 + r];
        sH[i + 8 * hf][n0 + r] = v > 0.f ? v : 0.f;
      }
    }
    __syncthreads();
    // GEMM2: g = tanh(h1 @ w2[:, gb:gb+128] + b2slice)
    for (int nt = 2 * wv; nt < 2 * wv + 2; ++nt) {
      int n0 = nt * 16;
      v8f c = {};
      for (int kk = 0; kk < 128; kk += 4) {
        v2f a; a.x = sH[r][kk + 2 * hf]; a.y = sH[r][kk + 2 * hf + 1];
        const float* bp = w2 + (size_t)(kk + 2 * hf) * 8192 + gb + n0 + r;
        v2f b; b.x = bp[0]; b.y = bp[8192];
        c = wmma_f32_4(a, b, c);
      }
#pragma unroll
      for (int i = 0; i < 8; ++i)
        sG[i + 8 * hf][n0 + r] = tanhf(c[i] + sb2[n0 + r]);
    }
    __syncthreads();
    // z[h][n] += sum_d g[n][h'*64+d] * xe[d]   (exclusive rows -> plain store)
    if (tid < 32) {
      int hp = tid >> 4, n = tid & 15;
      int hg = 2 * g + hp;
      float acc = 0.f;
      for (int d = 0; d < 64; ++d) acc += sG[n][hp * 64 + d] * sxe[d];
      zbuf[hg * 16 + n] = sz[hg][n] + acc;
    }
    __threadfence();
    gbarrier(bar, bar + 1);
    for (int i = tid; i < 2048; i += 128) sz[i >> 4][i & 15] = zbuf[i];
    __syncthreads();
    // da[n][m] = sum_h zta[n][m][h] * z[h][n]
    for (int idx = tid; idx < 256; idx += 128) {
      int i = idx >> 4, j = idx & 15;
      const float* zp = zta + i * 2048 + j * 128;
      float acc = 0.f;
      for (int h = 0; h < 128; ++h) acc += zp[h] * sz[h][i];
      sda[i][j] = acc;
    }
    __syncthreads();
    for (int idx = tid; idx < 512; idx += 128) {
      int i = idx >> 5, j = idx & 31;
      float acc = sab1[j];
      for (int m = 0; m < 16; ++m) acc += sda[i][m] * sw1a[m * 32 + j];
      shid[i][j] = acc > 0.f ? acc : 0.f;
    }
    __syncthreads();
    for (int idx = tid; idx < 256; idx += 128) {
      int i = idx >> 4, j = idx & 15;
      float acc = sab2[j];
      for (int m = 0; m < 32; ++m) acc += shid[i][m] * sw2a[m * 16 + j];
      sa[i][j] += tanhf(acc) * sde[j];
    }
    gbarrier(bar, bar + 1);  // nobody rewrites zbuf before all re-read it
  }
}

// ---------------------------------------------------------------------------
// K5: out = zf.flatten() @ dec_w + dec_b
// ---------------------------------------------------------------------------
__global__ __launch_bounds__(32) void k_dec(const float* __restrict__ zbuf,
                                            const float* __restrict__ dw,
                                            const float* __restrict__ db,
                                            float* __restrict__ out) {
  int o = threadIdx.x;
  if (o < 16) {
    float acc = db[o];
    for (int i = 0; i < 2048; ++i) acc += zbuf[i] * dw[i * 16 + o];
    out[o] = acc;
  }
}

extern "C" void kernel_launch(void* const* d_in, const int* in_sizes, int n_in,
                              void* d_out, int out_size, void* d_ws, size_t ws_size,
                              hipStream_t stream) {
  (void)in_sizes; (void)n_in; (void)out_size; (void)ws_size;
  const float* video   = (const float*)d_in[0];
  const float* bb_w1   = (const float*)d_in[1];
  const float* bb_b1   = (const float*)d_in[2];
  const float* bb_w2   = (const float*)d_in[3];
  const float* bb_b2   = (const float*)d_in[4];
  const float* key_w   = (const float*)d_in[5];
  const float* key_b   = (const float*)d_in[6];
  const float* query_w = (const float*)d_in[7];
  const float* query_b = (const float*)d_in[8];
  const float* value_w = (const float*)d_in[9];
  const float* value_b = (const float*)d_in[10];
  const float* ghe     = (const float*)d_in[11];
  const float* zta     = (const float*)d_in[12];
  const float* cde_w1  = (const float*)d_in[13];
  const float* cde_b1  = (const float*)d_in[14];
  const float* cde_w2  = (const float*)d_in[15];
  const float* cde_b2  = (const float*)d_in[16];
  const float* att_w1  = (const float*)d_in[17];
  const float* att_b1  = (const float*)d_in[18];
  const float* att_w2  = (const float*)d_in[19];
  const float* att_b2  = (const float*)d_in[20];
  const float* xd_w    = (const float*)d_in[21];
  const float* xd_b    = (const float*)d_in[22];
  const float* dec_w   = (const float*)d_in[23];
  const float* dec_b   = (const float*)d_in[24];

  float* ws    = (float*)d_ws;
  float* hpart = ws + WS_HPART;
  float* xg    = ws + WS_X;
  float* a0g   = ws + WS_A0;
  float* zb    = ws + WS_Z;
  float* evg   = ws + WS_EV;
  float* dag   = ws + WS_DATT;
  unsigned* bar = (unsigned*)(ws + WS_BAR);

  hipMemsetAsync(bar, 0, 2 * sizeof(unsigned), stream);
  k_bb<<<dim3(4, 8, NCHUNKS), 32, 0, stream>>>(video, bb_w1, hpart);
  k_mid<<<1, 256, 0, stream>>>(hpart, bb_b1, bb_w2, bb_b2, key_w, key_b,
                               query_w, query_b, value_w, value_b, ghe,
                               xg, a0g, zb);
  k_spline<<<1, 256, 0, stream>>>(xg, xd_w, xd_b, evg, dag);
  k_scan<<<NWG_SCAN, 128, 0, stream>>>(cde_w1, cde_b1, cde_w2, cde_b2, zta,
                                       att_w1, att_b1, att_w2, att_b2,
                                       evg, dag, a0g, zb, bar);
  k_dec<<<1, 32, 0, stream>>>(zb, dec_w, dec_b, (float*)d_out);
}